// RandomMaskSubgraphs_41283225649461
// MI455X (gfx1250) — compile-verified
//
#include <hip/hip_runtime.h>
#include <stdint.h>

// ---------------- constants from the reference ----------------
#define N_NODES   200000          // USER + ITEM
#define SAMP_NUM  40000           // int(N * KEEP_RATE)
#define BM_WORDS  8192            // >= ceil(200000/32) = 6250, padded
#define BT        256             // block threads (8 wave32s)
#define GSIZE     128             // edges per wave group (32 lanes x 4 via b128)
#define WTILE     2048            // edges per wave tile (16 groups)

typedef unsigned long long u64;
#define HS_EMPTY 0xFFFFFFFFFFFFFFFFull

// ---------------- small device helpers ----------------
static __device__ __forceinline__ u64 splitmix64(u64 x) {
    x += 0x9E3779B97F4A7C15ull;
    x = (x ^ (x >> 30)) * 0xBF58476D1CE4E5B9ull;
    x = (x ^ (x >> 27)) * 0x94D049BB133111EBull;
    return x ^ (x >> 31);
}

// returns true iff this call newly inserted the key
static __device__ __forceinline__ bool hs_insert(u64 key, u64* tab, uint32_t mask) {
    u64 h = key * 0x9E3779B97F4A7C15ull;
    uint32_t slot = (uint32_t)(h >> 32) & mask;
    for (uint32_t probe = 0; probe <= mask; ++probe) {
        u64 prev = atomicCAS(&tab[slot], HS_EMPTY, key);
        if (prev == HS_EMPTY) return true;
        if (prev == key) return false;
        slot = (slot + 1) & mask;
    }
    return false; // table full: key dropped (not in table, not counted)
}

// keep/hit classification for one edge
static __device__ __forceinline__ void classify(int r, int c,
                                                const uint32_t* __restrict__ bmHit,
                                                bool& keep, bool& hit) {
    bool valid = ((uint32_t)r < N_NODES) & ((uint32_t)c < N_NODES);
    hit = false;
    if (valid) {
        uint32_t h = (bmHit[(uint32_t)r >> 5] >> (r & 31)) |
                     (bmHit[(uint32_t)c >> 5] >> (c & 31));
        hit = (h & 1u) != 0u;
    }
    keep = valid && !hit;
}

static __device__ __forceinline__ void mark_hit(int r, int c,
                                                uint32_t* bmNext, uint32_t* bmMask) {
    atomicOr(&bmNext[(uint32_t)r >> 5], 1u << (r & 31));
    atomicOr(&bmNext[(uint32_t)c >> 5], 1u << (c & 31));
    atomicOr(&bmMask[(uint32_t)r >> 5], 1u << (r & 31));
    atomicOr(&bmMask[(uint32_t)c >> 5], 1u << (c & 31));
}

// wave32 inclusive scan
static __device__ __forceinline__ uint32_t wave_incl_scan(uint32_t v, int lane) {
    #pragma unroll
    for (int d = 1; d < 32; d <<= 1) {
        uint32_t t = __shfl_up(v, d, 32);
        if (lane >= d) v += t;
    }
    return v;
}
// wave32 sum reduce
static __device__ __forceinline__ uint32_t wave_reduce(uint32_t v) {
    #pragma unroll
    for (int d = 16; d; d >>= 1) v += __shfl_down(v, d, 32);
    return v;
}

// ---------------- fill / init kernels ----------------
__global__ void k_fill_u32(uint32_t* p, uint32_t v, int n) {
    for (int i = blockIdx.x * blockDim.x + threadIdx.x; i < n; i += gridDim.x * blockDim.x)
        p[i] = v;
}
// 16B-vector fill of u64 pattern (global_store_b128)
__global__ void k_fill_u64x2(ulonglong2* p, u64 v, int n2) {
    ulonglong2 q; q.x = v; q.y = v;
    for (int i = blockIdx.x * blockDim.x + threadIdx.x; i < n2; i += gridDim.x * blockDim.x)
        p[i] = q;
}
__global__ void k_fill_f32(float* p, float v, int n) {
    for (int i = blockIdx.x * blockDim.x + threadIdx.x; i < n; i += gridDim.x * blockDim.x)
        p[i] = v;
}
// 16B-vector zero of float buffer + scalar tail
__global__ void k_fill_f32v(float4* p, int n4, float* tail, int nTail) {
    float4 z; z.x = z.y = z.z = z.w = 0.0f;
    for (int i = blockIdx.x * blockDim.x + threadIdx.x; i < n4; i += gridDim.x * blockDim.x)
        p[i] = z;
    int t = blockIdx.x * blockDim.x + threadIdx.x;
    if (t < nTail) tail[t] = 0.0f;
}

// ---------------- marking kernels ----------------
__global__ void k_mark_seeds(const int* __restrict__ seeds, int n,
                             uint32_t* bmHit, uint32_t* bmMask) {
    for (int i = blockIdx.x * blockDim.x + threadIdx.x; i < n; i += gridDim.x * blockDim.x) {
        uint32_t s = (uint32_t)seeds[i];
        if (s >= N_NODES) continue;
        atomicOr(&bmHit[s >> 5], 1u << (s & 31));
        atomicOr(&bmMask[s >> 5], 1u << (s & 31));
    }
}
__global__ void k_mark_sampled(int n, uint32_t* bmMask) {
    for (int i = blockIdx.x * blockDim.x + threadIdx.x; i < n; i += gridDim.x * blockDim.x) {
        uint32_t s = (uint32_t)(splitmix64(0xA5A5A5A5ull + (u64)i) % N_NODES);
        atomicOr(&bmMask[s >> 5], 1u << (s & 31));
    }
}

// ---------------- deterministic stream compaction ----------------
// Phase 1: per-wave-tile keep counts (+ frontier marking on pass 0).
__global__ void __launch_bounds__(BT)
k_count_keep(const int* __restrict__ eR, const int* __restrict__ eC,
             int nConst, const uint32_t* __restrict__ nPtr,
             const uint32_t* __restrict__ bmHit,
             uint32_t* bmNext, uint32_t* bmMask, int markNext,
             uint32_t* __restrict__ tileCnt)
{
    const int n = nPtr ? (int)*nPtr : nConst;
    const int nTiles = (n + WTILE - 1) / WTILE;
    const int lane = threadIdx.x & 31;
    const int wave = (blockIdx.x * BT + threadIdx.x) >> 5;
    const int nWaves = (gridDim.x * BT) >> 5;
    const int4* vR = (const int4*)eR;
    const int4* vC = (const int4*)eC;

    for (int t = wave; t < nTiles; t += nWaves) {
        int tb = t * WTILE;
        int te = min(tb + WTILE, n);
        int ng = (te - tb) / GSIZE;
        uint32_t sum = 0;
        for (int g = 0; g < ng; ++g) {
            int e0 = tb + g * GSIZE + lane * 4;
            int4 r4 = vR[e0 >> 2];
            int4 c4 = vC[e0 >> 2];
            int rr[4] = {r4.x, r4.y, r4.z, r4.w};
            int cc[4] = {c4.x, c4.y, c4.z, c4.w};
            #pragma unroll
            for (int j = 0; j < 4; ++j) {
                bool k, h;
                classify(rr[j], cc[j], bmHit, k, h);
                sum += k;
                if (h && markNext) mark_hit(rr[j], cc[j], bmNext, bmMask);
            }
        }
        for (int i = tb + ng * GSIZE + lane; i < te; i += 32) {
            int r = eR[i], c = eC[i];
            bool k, h;
            classify(r, c, bmHit, k, h);
            sum += k;
            if (h && markNext) mark_hit(r, c, bmNext, bmMask);
        }
        sum = wave_reduce(sum);
        if (lane == 0) tileCnt[t] = sum;
    }
}

// Phase 2: single-block exclusive scan over m = ceil(n/divisor) counts; total -> *total.
__global__ void k_scan_excl(uint32_t* __restrict__ arr,
                            int nConst, const uint32_t* __restrict__ nPtr,
                            int divisor, uint32_t* __restrict__ total)
{
    __shared__ uint32_t s[256];
    const int n = nPtr ? (int)*nPtr : nConst;
    const int m = (n + divisor - 1) / divisor;
    uint32_t carry = 0;
    for (int chunk = 0; chunk < m; chunk += 256) {
        int i = chunk + (int)threadIdx.x;
        uint32_t v = (i < m) ? arr[i] : 0u;
        s[threadIdx.x] = v;
        __syncthreads();
        for (int d = 1; d < 256; d <<= 1) {
            uint32_t tv = (threadIdx.x >= (unsigned)d) ? s[threadIdx.x - d] : 0u;
            __syncthreads();
            s[threadIdx.x] += tv;
            __syncthreads();
        }
        if (i < m) arr[i] = carry + s[threadIdx.x] - v;
        uint32_t last = s[255];
        __syncthreads();
        carry += last;
    }
    if (threadIdx.x == 0) *total = carry;
}

// Phase 3: stable compaction write. Async global->LDS double-buffered stream
// (CDNA5 GLOBAL_LOAD_ASYNC_TO_LDS_B128 + S_WAIT_ASYNCCNT) per wave.
__global__ void __launch_bounds__(BT)
k_write_keep(const int* __restrict__ eR, const int* __restrict__ eC,
             int nConst, const uint32_t* __restrict__ nPtr,
             const uint32_t* __restrict__ bmHit,
             const uint32_t* __restrict__ tileOff,
             int* __restrict__ outR, int* __restrict__ outC)
{
    __shared__ __align__(16) int sR[2][BT * 4];
    __shared__ __align__(16) int sC[2][BT * 4];

    const int n = nPtr ? (int)*nPtr : nConst;
    const int nTiles = (n + WTILE - 1) / WTILE;
    const int lane = threadIdx.x & 31;
    const int wIB = threadIdx.x >> 5;
    const int wave = (blockIdx.x * BT + threadIdx.x) >> 5;
    const int nWaves = (gridDim.x * BT) >> 5;
    const int sl = wIB * GSIZE + lane * 4;

    // LDS byte addresses of this thread's 16B slot per buffer
    uint32_t ldsR[2], ldsC[2];
    ldsR[0] = (uint32_t)(uintptr_t)&sR[0][sl];
    ldsR[1] = (uint32_t)(uintptr_t)&sR[1][sl];
    ldsC[0] = (uint32_t)(uintptr_t)&sC[0][sl];
    ldsC[1] = (uint32_t)(uintptr_t)&sC[1][sl];

    for (int t = wave; t < nTiles; t += nWaves) {
        int tb = t * WTILE;
        int te = min(tb + WTILE, n);
        int ng = (te - tb) / GSIZE;
        uint32_t off = tileOff[t];

        if (ng > 0) {
            uint32_t voff = (uint32_t)(tb + lane * 4) * 4u;
            asm volatile("global_load_async_to_lds_b128 %0, %1, %2"
                         :: "v"(ldsR[0]), "v"(voff), "s"(eR) : "memory");
            asm volatile("global_load_async_to_lds_b128 %0, %1, %2"
                         :: "v"(ldsC[0]), "v"(voff), "s"(eC) : "memory");
        }
        for (int g = 0; g < ng; ++g) {
            if (g + 1 < ng) {
                uint32_t voff = (uint32_t)(tb + (g + 1) * GSIZE + lane * 4) * 4u;
                int nb = (g + 1) & 1;
                asm volatile("global_load_async_to_lds_b128 %0, %1, %2"
                             :: "v"(ldsR[nb]), "v"(voff), "s"(eR) : "memory");
                asm volatile("global_load_async_to_lds_b128 %0, %1, %2"
                             :: "v"(ldsC[nb]), "v"(voff), "s"(eC) : "memory");
                asm volatile("s_wait_asynccnt 2" ::: "memory"); // current group landed
            } else {
                asm volatile("s_wait_asynccnt 0" ::: "memory");
            }
            int b = g & 1;
            int rr[4], cc[4];
            #pragma unroll
            for (int j = 0; j < 4; ++j) { rr[j] = sR[b][sl + j]; cc[j] = sC[b][sl + j]; }
            uint32_t c = 0;
            bool kp[4];
            #pragma unroll
            for (int j = 0; j < 4; ++j) {
                bool k, h;
                classify(rr[j], cc[j], bmHit, k, h);
                kp[j] = k;
                c += k;
            }
            uint32_t incl = wave_incl_scan(c, lane);
            uint32_t tot = __shfl(incl, 31, 32);
            uint32_t p = off + (incl - c);
            #pragma unroll
            for (int j = 0; j < 4; ++j) {
                if (kp[j]) { outR[p] = rr[j]; outC[p] = cc[j]; ++p; }
            }
            off += tot;
        }
        // tail edges, stable order via ballot ranks
        for (int i0 = tb + ng * GSIZE; i0 < te; i0 += 32) {
            int i = i0 + lane;
            bool k = false;
            int r = 0, cv = 0;
            if (i < te) { r = eR[i]; cv = eC[i]; bool h; classify(r, cv, bmHit, k, h); }
            uint32_t mask = (uint32_t)__ballot(k);
            uint32_t rank = __popc(mask & ((1u << lane) - 1u));
            if (k) { outR[off + rank] = r; outC[off + rank] = cv; }
            off += __popc(mask);
        }
    }
}

// ---------------- sorted, deterministic mask-node list ----------------
__global__ void k_word_popc(const uint32_t* __restrict__ bm, uint32_t* __restrict__ wc) {
    for (int w = blockIdx.x * blockDim.x + threadIdx.x; w < BM_WORDS;
         w += gridDim.x * blockDim.x)
        wc[w] = (uint32_t)__popc(bm[w]);
}
__global__ void k_mask_write(const uint32_t* __restrict__ bm,
                             const uint32_t* __restrict__ wcOff, int* __restrict__ list) {
    for (int w = blockIdx.x * blockDim.x + threadIdx.x; w < BM_WORDS;
         w += gridDim.x * blockDim.x) {
        uint32_t bits = bm[w];
        uint32_t p = wcOff[w];
        while (bits) {
            int b = __ffs(bits) - 1;
            bits &= bits - 1;
            list[p++] = (w << 5) + b;
        }
    }
}

// ---------------- encoder normalization + output ----------------
__global__ void k_deg_accum(const int* __restrict__ br,
                            const uint32_t* __restrict__ cnt2, float* deg) {
    int n = (int)*cnt2;
    int stride = gridDim.x * blockDim.x;
    for (int i = blockIdx.x * blockDim.x + threadIdx.x; i < n; i += stride) {
        if (i + stride < n) __builtin_prefetch(&br[i + stride], 0, 1);
        atomicAdd(&deg[br[i]], 1.0f);
    }
}
__global__ void k_deg_finish(float* deg) {
    for (int i = blockIdx.x * blockDim.x + threadIdx.x; i < N_NODES;
         i += gridDim.x * blockDim.x)
        deg[i] = rsqrtf(deg[i] + 1e-12f);
}
__global__ void k_enc_write(const int* __restrict__ br, const int* __restrict__ bc,
                            const uint32_t* __restrict__ cnt2,
                            const float* __restrict__ deg,
                            float* __restrict__ out, int outSize) {
    int n = (int)*cnt2;
    for (int i = blockIdx.x * blockDim.x + threadIdx.x; i < n;
         i += gridDim.x * blockDim.x) {
        int r = br[i], c = bc[i];
        size_t i2 = 2ull * (size_t)n + i;
        if (i2 < (size_t)outSize) {
            out[i] = (float)r;
            out[(size_t)n + i] = (float)c;
            out[i2] = deg[r] * deg[c];
        }
    }
}

// ---------------- decoder hash-set + output ----------------
// All insert kernels count NEW insertions (wave-aggregated) so no separate
// counting pass over the table is needed.
__global__ void k_ins_self(u64* tab, uint32_t mask, uint32_t* cntDec) {
    const int lane = threadIdx.x & 31;
    uint32_t nw = 0;
    for (int i = blockIdx.x * blockDim.x + threadIdx.x; i < N_NODES;
         i += gridDim.x * blockDim.x)
        nw += hs_insert((u64)i * N_NODES + (u64)i, tab, mask);
    nw = wave_reduce(nw);
    if (lane == 0 && nw) atomicAdd(cntDec, nw);
}
__global__ void k_ins_kept(const int* __restrict__ br, const int* __restrict__ bc,
                           const uint32_t* __restrict__ cnt2, u64* tab, uint32_t mask,
                           uint32_t* cntDec) {
    const int lane = threadIdx.x & 31;
    int n = (int)*cnt2;
    uint32_t nw = 0;
    for (int i = blockIdx.x * blockDim.x + threadIdx.x; i < n;
         i += gridDim.x * blockDim.x)
        nw += hs_insert((u64)br[i] * N_NODES + (u64)bc[i], tab, mask);
    nw = wave_reduce(nw);
    if (lane == 0 && nw) atomicAdd(cntDec, nw);
}
__global__ void k_ins_tem(const int* __restrict__ maskList,
                          const uint32_t* __restrict__ cntM,
                          int nnz, u64* tab, uint32_t mask, uint32_t* cntDec) {
    const int lane = threadIdx.x & 31;
    uint32_t tm = *cntM;
    if (tm == 0) return;
    int stride = gridDim.x * blockDim.x;
    uint32_t nw = 0;
    for (int i = blockIdx.x * blockDim.x + threadIdx.x; i < nnz; i += stride) {
        u64 h1 = splitmix64(0x1111000000000000ull + (u64)i);
        u64 h2 = splitmix64(0x2222000000000000ull + (u64)i);
        int r = maskList[(uint32_t)(h1 % tm)];
        int c = maskList[(uint32_t)(h2 % tm)];
        nw += hs_insert((u64)r * N_NODES + (u64)c, tab, mask);
        nw += hs_insert((u64)c * N_NODES + (u64)r, tab, mask);
    }
    nw = wave_reduce(nw);
    if (lane == 0 && nw) atomicAdd(cntDec, nw);
}
// Wave-aggregated compaction of occupied slots (one atomic per wave32).
__global__ void k_dec_write(const u64* __restrict__ tab, int slots,
                            const uint32_t* __restrict__ cnt2,
                            const uint32_t* __restrict__ cntDec, uint32_t* cntPos,
                            float* __restrict__ out, int outSize) {
    const int lane = threadIdx.x & 31;
    size_t base = 3ull * (size_t)(*cnt2);
    size_t E2 = (size_t)(*cntDec);
    int gid = blockIdx.x * blockDim.x + threadIdx.x;
    int stride = gridDim.x * blockDim.x;
    for (int b0 = gid - lane; b0 < slots; b0 += stride) {
        int idx = b0 + lane;
        u64 key = (idx < slots) ? tab[idx] : HS_EMPTY;
        bool occ = (key != HS_EMPTY);
        uint32_t mask = (uint32_t)__ballot(occ);
        uint32_t cw = __popc(mask);
        uint32_t basep = 0;
        if (lane == 0 && cw) basep = atomicAdd(cntPos, cw);
        basep = __shfl(basep, 0, 32);
        if (occ) {
            uint32_t p = basep + __popc(mask & ((1u << lane) - 1u));
            size_t ir = base + p, ic = base + E2 + p, iv = base + 2 * E2 + p;
            if (iv < (size_t)outSize) {
                out[ir] = (float)(key / N_NODES);
                out[ic] = (float)(key % N_NODES);
                out[iv] = 1.0f;
            }
        }
    }
}

// ---------------- host orchestration ----------------
extern "C" void kernel_launch(void* const* d_in, const int* in_sizes, int n_in,
                              void* d_out, int out_size, void* d_ws, size_t ws_size,
                              hipStream_t stream) {
    const int* eR    = (const int*)d_in[0];
    const int* eC    = (const int*)d_in[1];
    /* d_in[2] = adj_vals (unused: encoder uses unit values) */
    const int* seeds = (const int*)d_in[3];
    const int nE     = in_sizes[0];
    const int nSeeds = in_sizes[3];
    float* out = (float*)d_out;
    (void)n_in;

    const int nTilesMax = (nE + WTILE - 1) / WTILE;
    const int tcWords = (nTilesMax > BM_WORDS) ? nTilesMax : BM_WORDS;

    // --- workspace layout (256B-aligned slabs) ---
    char* ws = (char*)d_ws;
    size_t cur = 0;
    auto take = [&](size_t bytes) -> size_t {
        size_t o = cur;
        cur += (bytes + 255) & ~(size_t)255;
        return o;
    };
    size_t o_cnt = take(256);                 // [0]=E(after d0) [1]=E1 [2]=cntM [3]=E2 [4]=pos
    size_t o_bmH = take((size_t)BM_WORDS * 4);
    size_t o_bmN = take((size_t)BM_WORDS * 4);
    size_t o_bmM = take((size_t)BM_WORDS * 4);
    size_t o_deg = take((size_t)N_NODES * 4);
    size_t o_ml  = take((size_t)N_NODES * 4);
    size_t o_tc  = take((size_t)tcWords * 4); // tile counts / word counts (reused)
    size_t o_ar  = take((size_t)nE * 4);
    size_t o_ac  = take((size_t)nE * 4);
    size_t o_br  = take((size_t)nE * 4);
    size_t o_bc  = take((size_t)nE * 4);
    size_t remain = (ws_size > cur) ? (ws_size - cur) : 0;
    size_t slots = 1ull << 16;
    while (slots * 2 * 8 <= remain && slots < (1ull << 26)) slots <<= 1;
    size_t o_tab = take(slots * 8);

    uint32_t* cnt    = (uint32_t*)(ws + o_cnt);
    uint32_t* bmHit  = (uint32_t*)(ws + o_bmH);
    uint32_t* bmNext = (uint32_t*)(ws + o_bmN);
    uint32_t* bmMask = (uint32_t*)(ws + o_bmM);
    float*    deg    = (float*)(ws + o_deg);
    int*      maskL  = (int*)(ws + o_ml);
    uint32_t* tc     = (uint32_t*)(ws + o_tc);
    int*      aR     = (int*)(ws + o_ar);
    int*      aC     = (int*)(ws + o_ac);
    int*      bR     = (int*)(ws + o_br);
    int*      bC     = (int*)(ws + o_bc);
    u64*      tab    = (u64*)(ws + o_tab);
    uint32_t  tmask  = (uint32_t)(slots - 1);

    auto blocks = [](long long n, int bt) {
        long long b = (n + bt - 1) / bt;
        if (b > 4096) b = 4096;
        if (b < 1) b = 1;
        return (int)b;
    };
    int bCW = (int)((nTilesMax + 7) / 8);  // 8 waves per 256-thread block
    if (bCW > 2048) bCW = 2048;
    if (bCW < 1) bCW = 1;

    // --- init (every call: harness does not re-poison between replays) ---
    int initWords = (int)((o_deg - o_cnt) / 4); // counters + 3 bitmaps, contiguous
    k_fill_u32<<<blocks(initWords, 256), 256, 0, stream>>>((uint32_t*)(ws + o_cnt), 0u, initWords);
    k_fill_f32<<<blocks(N_NODES, 256), 256, 0, stream>>>(deg, 0.0f, N_NODES);
    k_fill_u64x2<<<blocks((long long)(slots / 2), 256), 256, 0, stream>>>(
        (ulonglong2*)tab, HS_EMPTY, (int)(slots / 2));
    {
        int n4 = out_size / 4;
        int nTail = out_size - n4 * 4;
        k_fill_f32v<<<blocks(n4 > 0 ? n4 : 1, 256), 256, 0, stream>>>(
            (float4*)out, n4, out + (size_t)n4 * 4, nTail);
    }

    // --- depth-0: mark seeds/sampled, count+mark, scan, stable write ---
    k_mark_seeds<<<blocks(nSeeds, 256), 256, 0, stream>>>(seeds, nSeeds, bmHit, bmMask);
    k_mark_sampled<<<blocks(SAMP_NUM, 256), 256, 0, stream>>>(SAMP_NUM, bmMask);
    k_count_keep<<<bCW, BT, 0, stream>>>(eR, eC, nE, nullptr, bmHit,
                                         bmNext, bmMask, /*markNext=*/1, tc);
    k_scan_excl<<<1, 256, 0, stream>>>(tc, nE, nullptr, WTILE, &cnt[0]);
    k_write_keep<<<bCW, BT, 0, stream>>>(eR, eC, nE, nullptr, bmHit, tc, aR, aC);

    // --- depth-1: filter survivors against the frontier bitmap ---
    k_count_keep<<<bCW, BT, 0, stream>>>(aR, aC, 0, &cnt[0], bmNext,
                                         bmNext, bmMask, /*markNext=*/0, tc);
    k_scan_excl<<<1, 256, 0, stream>>>(tc, 0, &cnt[0], WTILE, &cnt[1]);
    k_write_keep<<<bCW, BT, 0, stream>>>(aR, aC, 0, &cnt[0], bmNext, tc, bR, bC);

    // --- sorted mask-node list (union of seeds, frontier, sampled) ---
    k_word_popc<<<blocks(BM_WORDS, 256), 256, 0, stream>>>(bmMask, tc);
    k_scan_excl<<<1, 256, 0, stream>>>(tc, BM_WORDS, nullptr, 1, &cnt[2]);
    k_mask_write<<<blocks(BM_WORDS, 256), 256, 0, stream>>>(bmMask, tc, maskL);

    // --- encoder normalization + output (order matches reference) ---
    k_deg_accum<<<blocks(nE, 256), 256, 0, stream>>>(bR, &cnt[1], deg);
    k_deg_finish<<<blocks(N_NODES, 256), 256, 0, stream>>>(deg);
    k_enc_write<<<blocks(nE, 256), 256, 0, stream>>>(bR, bC, &cnt[1], deg, out, out_size);

    // --- decoder edge set: self loops + kept edges + rewiring, dedup;
    //     E2 counted at insert time (no extra table pass) ---
    k_ins_self<<<blocks(N_NODES, 256), 256, 0, stream>>>(tab, tmask, &cnt[3]);
    k_ins_kept<<<blocks(nE, 256), 256, 0, stream>>>(bR, bC, &cnt[1], tab, tmask, &cnt[3]);
    k_ins_tem<<<blocks(nE, 256), 256, 0, stream>>>(maskL, &cnt[2], nE, tab, tmask, &cnt[3]);

    // --- decoder output ---
    k_dec_write<<<blocks((long long)slots, 256), 256, 0, stream>>>(
        tab, (int)slots, &cnt[1], &cnt[3], &cnt[4], out, out_size);
}